// SphericalBasisLayer_46248207843576
// MI455X (gfx1250) — compile-verified
//
#include <hip/hip_runtime.h>
#include <cmath>

#define NUM_L 7
#define NUM_N 6
#define INV_CUTOFF 0.2f

typedef float v2f __attribute__((ext_vector_type(2)));
typedef float v8f __attribute__((ext_vector_type(8)));

struct Tables {
    float zs[NUM_L][NUM_N];   // bessel zeros z_{l,n}
    float sc[NUM_L][NUM_N];   // sqrt(2)/|j_{l+1}(z)| * cutoff^-1.5 (prefused)
};

// Legendre-monomial coefficients scaled by sqrt((2l+1)/(4*pi)); rows 7..15 zero (pad for WMMA M=16)
__constant__ float COEF[16][8] = {
    { 0.28209479177387814f, 0.f, 0.f, 0.f, 0.f, 0.f, 0.f, 0.f},
    { 0.f, 0.48860251190291992f, 0.f, 0.f, 0.f, 0.f, 0.f, 0.f},
    {-0.31539156525252005f, 0.f, 0.94617469575756008f, 0.f, 0.f, 0.f, 0.f, 0.f},
    { 0.f, -1.11952899777035462f, 0.f, 1.86588166295059103f, 0.f, 0.f, 0.f, 0.f},
    { 0.31735664074561294f, 0.f, -3.17356640745612940f, 0.f, 3.70249414203215080f, 0.f, 0.f, 0.f},
    { 0.f, 1.75425483680135420f, 0.f, -8.18652257173965300f, 0.f, 7.36787031456568700f, 0.f, 0.f},
    {-0.31784601133814210f, 0.f, 6.67476623810098500f, 0.f, -20.02429871430295400f, 0.f, 14.68448572382216600f, 0.f},
    {0.f,0.f,0.f,0.f,0.f,0.f,0.f,0.f}, {0.f,0.f,0.f,0.f,0.f,0.f,0.f,0.f},
    {0.f,0.f,0.f,0.f,0.f,0.f,0.f,0.f}, {0.f,0.f,0.f,0.f,0.f,0.f,0.f,0.f},
    {0.f,0.f,0.f,0.f,0.f,0.f,0.f,0.f}, {0.f,0.f,0.f,0.f,0.f,0.f,0.f,0.f},
    {0.f,0.f,0.f,0.f,0.f,0.f,0.f,0.f}, {0.f,0.f,0.f,0.f,0.f,0.f,0.f,0.f}
};

// ---- per-edge radial basis: envelope(d) * norm * j_l(d * z_{l,n}) ----
__device__ __forceinline__ void compute_rbf42(float ds, const Tables& t, float* __restrict__ vals) {
    float d5 = ds * ds; d5 = d5 * d5 * ds;                              // ds^5
    float env = 1.0f + d5 * (-21.0f + ds * (35.0f + ds * (-15.0f)));
    env = (ds < 1.0f) ? env : 0.0f;
#pragma unroll
    for (int l = 0; l < NUM_L; ++l) {
#pragma unroll
        for (int n = 0; n < NUM_N; ++n) {
            float x = ds * t.zs[l][n];
            float s, c;
            sincosf(x, &s, &c);
            float inv = 1.0f / x;
            float j = s * inv;                                          // j0
            if (l > 0) {
                float j1 = (s * inv - c) * inv;                         // j1
                float jm = j, jc = j1;
#pragma unroll
                for (int ll = 1; ll < l; ++ll) {
                    float tn = (2.0f * ll + 1.0f) * inv * jc - jm;
                    jm = jc; jc = tn;
                }
                j = jc;
            }
            vals[l * NUM_N + n] = env * t.sc[l][n] * j;
        }
    }
}

// ---- pass 1: one thread per edge -> rbf_env table in workspace (E x 42 f32, rows 8B-aligned) ----
__global__ void sbl_edge_kernel(const float* __restrict__ dist,
                                float* __restrict__ rbf, int E, Tables t) {
    int e = blockIdx.x * blockDim.x + threadIdx.x;
    if (e >= E) return;
    float vals[42];
    compute_rbf42(dist[e] * INV_CUTOFF, t, vals);
    v2f* rp = (v2f*)(rbf + (size_t)e * 42);          // e*42 even -> 8B aligned
#pragma unroll
    for (int k = 0; k < 21; ++k) {
        v2f v; v.x = vals[2 * k]; v.y = vals[2 * k + 1];
        rp[k] = v;
    }
}

// ---- pass 2: one wave per 16 angles. Y_l0 via V_WMMA_F32_16X16X4_F32:
//      sph(16 l-rows x 16 angles) = COEF(16x8) x [ct^k](8x16), K=8 as 2 chained K=4 WMMAs ----
__global__ void __launch_bounds__(128)
sbl_angle_kernel(const float* __restrict__ theta,
                 const int* __restrict__ nbr,
                 const float* __restrict__ rbf,
                 float* __restrict__ out, int A) {
    __shared__ float    sphS[4][16][8];
    __shared__ unsigned rowS[4][16];                  // e * 168 (byte offset of gather row)

    const int lane = threadIdx.x & 31;
    const int wid  = threadIdx.x >> 5;
    const int m    = lane & 15;                       // angle-in-tile (and coef row)
    const bool hi  = lane >= 16;
    const long aBase = ((long)blockIdx.x * 4 + wid) * 16;

    // both half-waves load the same angle m (L0 broadcast); clamp, never branch (EXEC must be full for WMMA)
    long aMy = aBase + m;
    long aCl = aMy < A ? aMy : (long)(A - 1);
    float ct = cosf(theta[aCl]);
    if (!hi) rowS[wid][m] = (unsigned)nbr[2 * aCl] * 168u;

    float c2 = ct * ct, c3 = c2 * ct, c4 = c2 * c2;
    float c5 = c4 * ct, c6 = c4 * c2, c7 = c4 * c3;

    // A-matrix 16x4 layout: lane m holds K = kb, kb+1 (kb = 0 for lanes 0-15, 2 for 16-31)
    // B-matrix 4x16 layout mirrored: lane holds rows kb, kb+1 at column m
    const int kb = hi ? 2 : 0;
    v2f a0, a1, b0, b1;
    a0.x = COEF[m][kb];     a0.y = COEF[m][kb + 1];
    a1.x = COEF[m][4 + kb]; a1.y = COEF[m][5 + kb];
    b0.x = hi ? c2 : 1.0f;  b0.y = hi ? c3 : ct;
    b1.x = hi ? c6 : c4;    b1.y = hi ? c7 : c5;

    v8f acc = {0.f, 0.f, 0.f, 0.f, 0.f, 0.f, 0.f, 0.f};
    acc = __builtin_amdgcn_wmma_f32_16x16x4_f32(false, a0, false, b0, (short)0, acc, false, false);
    acc = __builtin_amdgcn_wmma_f32_16x16x4_f32(false, a1, false, b1, (short)0, acc, false, false);

    // D layout: VGPR r, lanes 0-15 -> (M=r, N=lane) = sph[l=r][angle=lane]
    if (!hi) {
#pragma unroll
        for (int r = 0; r < 7; ++r) sphS[wid][m][r] = acc[r];
    }
    __syncthreads();

    if (aBase + 16 <= (long)A) {
        // ---- fast path: full tile. 16 angles * 42 out = 336 b64 pairs = 10.5 iters * 32 lanes.
        // Even pair offsets never straddle an l-boundary (42 % 6 == 0, pair starts even).
        float* outP = out + (size_t)aBase * 42 + 2 * lane;  // + it*64 folds into store immediate
        int a = (lane >= 21) ? 1 : 0;
        int r = 2 * lane - 42 * a;                          // element offset in row, even
        auto body = [&](int it) {
            unsigned rowb = rowS[wid][a];
            float    s    = sphS[wid][a][(unsigned)r / 6u];
            v2f      rv   = *(const v2f*)((const char*)rbf + (rowb + ((unsigned)r << 2)));
            v2f      ov;  ov.x = s * rv.x; ov.y = s * rv.y;
            __builtin_nontemporal_store(ov, (v2f*)(outP + it * 64));
        };
#pragma unroll
        for (int it = 0; it < 10; ++it) {
            body(it);
            r += 64;
            if (r >= 42) { r -= 42; ++a; }
            if (r >= 42) { r -= 42; ++a; }
        }
        if (lane < 16) body(10);                            // pairs 320..335
    } else {
        // ---- tail path (partial last tile, or a wave with aBase >= A): scalar, guarded ----
        int aL  = 0;
        int r42 = lane;
#pragma unroll
        for (int it = 0; it < 21; ++it) {
            long aG = aBase + aL;
            if (aG < A) {
                float    s    = sphS[wid][aL][(unsigned)r42 / 6u];
                unsigned rowb = rowS[wid][aL];
                float    v    = *(const float*)((const char*)rbf + (rowb + ((unsigned)r42 << 2)));
                out[aG * 42 + r42] = s * v;
            }
            r42 += 32;
            if (r42 >= 42) { r42 -= 42; ++aL; }
        }
    }
}

// ---- fallback if workspace too small: fully fused per-angle recompute ----
__global__ void sbl_fused_kernel(const float* __restrict__ dist,
                                 const float* __restrict__ theta,
                                 const int* __restrict__ nbr,
                                 float* __restrict__ out, int A, Tables t) {
    long a = (long)blockIdx.x * blockDim.x + threadIdx.x;
    if (a >= A) return;
    int e = nbr[2 * a];
    float vals[42];
    compute_rbf42(dist[e] * INV_CUTOFF, t, vals);
    float ct = cosf(theta[a]);
    float p[8];
    p[0] = 1.0f;
#pragma unroll
    for (int k = 1; k < 8; ++k) p[k] = p[k - 1] * ct;
    long base = a * 42;
#pragma unroll
    for (int l = 0; l < NUM_L; ++l) {
        float s = 0.0f;
#pragma unroll
        for (int k = 0; k < 8; ++k) s += COEF[l][k] * p[k];
#pragma unroll
        for (int n = 0; n < NUM_N; ++n) out[base + l * NUM_N + n] = s * vals[l * NUM_N + n];
    }
}

// ---- host: spherical bessel zeros (double) + normalizers ----
static const double ZD[NUM_L][NUM_N] = {
    { 3.141592653589793,  6.283185307179586,  9.424777960769380, 12.566370614359172, 15.707963267948966, 18.849555921538759},
    { 4.493409457909064,  7.725251836937707, 10.904121659428899, 14.066193912831473, 17.220755271930768, 20.371302959287561},
    { 5.763459196894550,  9.095011330476355, 12.322940970566582, 15.514603010886749, 18.689036355362822, 21.853874222709714},
    { 6.987932000500519, 10.417118547379365, 13.698023153249251, 16.923621285214318, 20.121806174454734, 23.304246988939651},
    { 8.182561452571243, 11.704907154570450, 15.039664707616340, 18.301255959541976, 21.525417733399924, 24.727565547835619},
    { 9.355812111042747, 12.966530172775317, 16.354709639348330, 19.653152101821083, 22.904550647902713, 26.127750137225693},
    {10.512835464101115, 14.207392458842461, 17.647974592752651, 20.983463068944967, 24.262768043575559, 27.507868364904542}
};

static double jl_host(int l, double x) {
    double j0 = sin(x) / x;
    if (l == 0) return j0;
    double j1 = sin(x) / (x * x) - cos(x) / x;
    double jm = j0, jc = j1;
    for (int ll = 1; ll < l; ++ll) { double tn = (2.0 * ll + 1.0) / x * jc - jm; jm = jc; jc = tn; }
    return jc;
}

extern "C" void kernel_launch(void* const* d_in, const int* in_sizes, int n_in,
                              void* d_out, int out_size, void* d_ws, size_t ws_size,
                              hipStream_t stream) {
    const float* dist  = (const float*)d_in[0];
    const float* theta = (const float*)d_in[1];
    const int*   nbr   = (const int*)d_in[2];
    float*       out   = (float*)d_out;

    const int E = in_sizes[0];
    const int A = in_sizes[1];

    Tables t;
    const double norm_const = pow(0.2, 1.5);   // cutoff^-1.5
    for (int l = 0; l < NUM_L; ++l)
        for (int n = 0; n < NUM_N; ++n) {
            double z = ZD[l][n];
            t.zs[l][n] = (float)z;
            t.sc[l][n] = (float)(sqrt(2.0) / fabs(jl_host(l + 1, z)) * norm_const);
        }

    const size_t needWs = (size_t)E * 42 * sizeof(float);
    if (ws_size >= needWs) {
        float* rbf = (float*)d_ws;
        sbl_edge_kernel<<<(E + 255) / 256, 256, 0, stream>>>(dist, rbf, E, t);
        long nTiles = ((long)A + 15) / 16;                 // waves
        int  blocks = (int)((nTiles + 3) / 4);             // 4 waves / block
        sbl_angle_kernel<<<blocks, 128, 0, stream>>>(theta, nbr, rbf, out, A);
    } else {
        sbl_fused_kernel<<<(int)(((long)A + 255) / 256), 256, 0, stream>>>(dist, theta, nbr, out, A, t);
    }
    (void)n_in; (void)out_size;
}